// SlotMatchingModule_51488067944939
// MI455X (gfx1250) — compile-verified
//
#include <hip/hip_runtime.h>

// SlotMatchingModule for MI455X (gfx1250):
//   out[i,j] = same-type ? dot_{k<1024}(ft1[i,k], ft2[j,k])
//                        : dot_{c<128}(ft1[i, t1[i]*128+c], ft2[j, t2[j]*128+c])
// fp32 WMMA (v_wmma_f32_16x16x4_f32) + double-buffered async global->LDS staging.

#define DIMK   1024
#define NCLS   128
#define BM     128
#define BN     128
#define BK     32
#define LDSTR  36   // padded LDS row stride (floats); multiple of 4 keeps 16B alignment

typedef float v2f __attribute__((ext_vector_type(2)));
typedef float v8f __attribute__((ext_vector_type(8)));

// ---- CDNA5 async copy primitives (ISA 10.6 / 15.18.3, ASYNCcnt-tracked) ----
__device__ __forceinline__ void async_copy_b128(unsigned lds_byte_addr,
                                                const float* gptr) {
    asm volatile("global_load_async_to_lds_b128 %0, %1, off"
                 :
                 : "v"(lds_byte_addr), "v"((unsigned long long)(uintptr_t)gptr)
                 : "memory");
}
__device__ __forceinline__ void wait_async0() {
    asm volatile("s_wait_asynccnt 0" ::: "memory");
}
// LDS aperture flat addresses carry the LDS byte offset in the low 32 bits.
__device__ __forceinline__ unsigned lds_off(const void* p) {
    return (unsigned)(uintptr_t)p;
}

__global__ __launch_bounds__(256)
void slot_matching_kernel(const float* __restrict__ ft1,
                          const float* __restrict__ ft2,
                          const int*   __restrict__ type1,
                          const int*   __restrict__ type2,
                          float* __restrict__ out,
                          int n2)
{
    __shared__ float sA[2][BM * LDSTR];
    __shared__ float sB[2][BN * LDSTR];
    __shared__ int   sT1[BM];
    __shared__ int   sT2[BN];

    const int tid     = threadIdx.x;
    const int lane    = tid & 31;
    const int wave    = tid >> 5;        // 0..7
    const int waveRow = wave >> 1;       // 0..3  -> 32-row strip
    const int waveCol = wave & 1;        // 0..1  -> 64-col strip
    const int wm0     = waveRow * 32;
    const int wn0     = waveCol * 64;
    const int half    = lane >> 4;       // 0: K pair {0,1}, 1: K pair {2,3}
    const int l15     = lane & 15;

    const int bm0 = blockIdx.y * BM;
    const int bn0 = blockIdx.x * BN;

    // Cache this block's row/col type ids.
    if (tid < BM)            sT1[tid]      = type1[bm0 + tid];
    else if (tid < BM + BN)  sT2[tid - BM] = type2[bn0 + (tid - BM)];
    __syncthreads();

    v8f accF[2][4];   // full-K accumulators
    v8f accX[2][4];   // cross (slot-slice) accumulators
    #pragma unroll
    for (int i = 0; i < 2; ++i)
        #pragma unroll
        for (int j = 0; j < 4; ++j) { accF[i][j] = (v8f){}; accX[i][j] = (v8f){}; }

    // Issue one chunk's worth of async global->LDS copies (cooperative, all 256 thr).
    auto issue_chunk = [&](int buf, int kk, bool gather) {
        #pragma unroll
        for (int i = 0; i < 4; ++i) {
            const int idx = tid + 256 * i;        // 0..1023
            const int r   = idx >> 3;             // tile row
            const int c4  = (idx & 7) << 2;       // float4 column
            size_t offA = (size_t)(bm0 + r) * DIMK + kk + c4;
            size_t offB = (size_t)(bn0 + r) * DIMK + kk + c4;
            if (gather) {
                offA += (size_t)sT1[r] * NCLS;
                offB += (size_t)sT2[r] * NCLS;
            }
            async_copy_b128(lds_off(&sA[buf][r * LDSTR + c4]), ft1 + offA);
            async_copy_b128(lds_off(&sB[buf][r * LDSTR + c4]), ft2 + offB);
        }
    };

    // One BK=32 chunk of WMMA work from LDS buffer `buf` into `acc`.
    auto compute_chunk = [&](int buf, v8f (&acc)[2][4]) {
        #pragma unroll
        for (int ks = 0; ks < BK; ks += 4) {
            v2f af[2], bf[4];
            #pragma unroll
            for (int tm = 0; tm < 2; ++tm)
                af[tm] = *(const v2f*)(&sA[buf][(wm0 + tm * 16 + l15) * LDSTR + ks + half * 2]);
            #pragma unroll
            for (int tn = 0; tn < 4; ++tn)
                bf[tn] = *(const v2f*)(&sB[buf][(wn0 + tn * 16 + l15) * LDSTR + ks + half * 2]);
            #pragma unroll
            for (int tm = 0; tm < 2; ++tm)
                #pragma unroll
                for (int tn = 0; tn < 4; ++tn)
                    acc[tm][tn] = __builtin_amdgcn_wmma_f32_16x16x4_f32(
                        false, af[tm], false, bf[tn], (short)0, acc[tm][tn], false, false);
        }
    };

    // ---------------- Phase 1: cross GEMM (K = 128, gathered slot slices) ----
    issue_chunk(0, 0, true);
    #pragma unroll 1
    for (int kk = 0; kk < NCLS; kk += BK) {
        const int cur = (kk / BK) & 1;
        wait_async0();
        __syncthreads();
        if (kk + BK < NCLS) issue_chunk(cur ^ 1, kk + BK, true);
        compute_chunk(cur, accX);
    }

    // ---------------- Phase 2: full GEMM (K = 1024) --------------------------
    __syncthreads();
    issue_chunk(0, 0, false);
    #pragma unroll 1
    for (int kk = 0; kk < DIMK; kk += BK) {
        const int cur = (kk / BK) & 1;
        wait_async0();
        __syncthreads();
        if (kk + BK < DIMK) issue_chunk(cur ^ 1, kk + BK, false);
        compute_chunk(cur, accF);
    }

    // ---------------- Store with per-element select (non-temporal) -----------
    #pragma unroll
    for (int tm = 0; tm < 2; ++tm) {
        #pragma unroll
        for (int tn = 0; tn < 4; ++tn) {
            #pragma unroll
            for (int r = 0; r < 8; ++r) {
                const int lr = wm0 + tm * 16 + r + half * 8;   // local row 0..127
                const int lc = wn0 + tn * 16 + l15;            // local col 0..127
                const bool same = (sT1[lr] == sT2[lc]);
                const float v = same ? accF[tm][tn][r] : accX[tm][tn][r];
                __builtin_nontemporal_store(v, &out[(size_t)(bm0 + lr) * (size_t)n2 + (bn0 + lc)]);
            }
        }
    }
}

extern "C" void kernel_launch(void* const* d_in, const int* in_sizes, int n_in,
                              void* d_out, int out_size, void* d_ws, size_t ws_size,
                              hipStream_t stream) {
    const float* ft1 = (const float*)d_in[0];
    const float* ft2 = (const float*)d_in[1];
    const int*   t1  = (const int*)d_in[2];
    const int*   t2  = (const int*)d_in[3];
    float* out = (float*)d_out;

    const int n1 = in_sizes[0] / DIMK;   // 8192
    const int n2 = in_sizes[1] / DIMK;   // 8192

    dim3 grid(n2 / BN, n1 / BM);
    dim3 block(256);
    slot_matching_kernel<<<grid, block, 0, stream>>>(ft1, ft2, t1, t2, out, n2);
}